// recur_module_24507083391506
// MI455X (gfx1250) — compile-verified
//
#include <hip/hip_runtime.h>

// out = x - max(ceil(x), 1.0f)  -- pure streaming elementwise map.
// Roofline: 1 GiB one-touch HBM traffic, ~46 us floor @ 23.3 TB/s; compute is
// ~3 VALU ops/element and free. Optimal kernel = b128 non-temporal load/store
// with minimal address overhead.
//
// Fast path: 32-bit indexing (byte extent < 2^31), exact-size grid, no loop:
// one global_load_b128 NT + 12 VALU + one global_store_b128 NT per thread,
// with GVS addressing (SGPR64 base + VGPR32 offset) instead of 64-bit vector
// address arithmetic.

typedef __attribute__((ext_vector_type(4))) float float4v;

__device__ __forceinline__ float4v ceil_route4(float4v v) {
  float4v r;
#pragma unroll
  for (int k = 0; k < 4; ++k) {
    float x = v[k];
    r[k] = x - __builtin_fmaxf(__builtin_ceilf(x), 1.0f);  // v_ceil/v_max_num/v_sub
  }
  return r;
}

// Fast path: one float4 per thread, 32-bit offsets, no loop.
__global__ __launch_bounds__(256) void ceil_route_vec4_fast(
    const float4v* __restrict__ in, float4v* __restrict__ out, int n4) {
  int i = (int)(blockIdx.x * blockDim.x + threadIdx.x);
  if (i < n4) {
    float4v v = __builtin_nontemporal_load(in + i);   // global_load_b128 NT
    __builtin_nontemporal_store(ceil_route4(v), out + i);  // global_store_b128 NT
  }
}

// Generic fallback: 64-bit grid-stride (only used for > 2^31-byte extents).
__global__ __launch_bounds__(256) void ceil_route_vec4_generic(
    const float4v* __restrict__ in, float4v* __restrict__ out, long long n4) {
  const long long stride = (long long)gridDim.x * (long long)blockDim.x;
  for (long long i = (long long)blockIdx.x * blockDim.x + threadIdx.x; i < n4;
       i += stride) {
    float4v v = __builtin_nontemporal_load(in + i);
    __builtin_nontemporal_store(ceil_route4(v), out + i);
  }
}

// Scalar tail for element counts not divisible by 4.
__global__ __launch_bounds__(64) void ceil_route_tail(
    const float* __restrict__ in, float* __restrict__ out, long long base,
    long long n) {
  long long i = base + (long long)blockIdx.x * blockDim.x + threadIdx.x;
  if (i < n) {
    float x = __builtin_nontemporal_load(in + i);
    __builtin_nontemporal_store(x - __builtin_fmaxf(__builtin_ceilf(x), 1.0f),
                                out + i);
  }
}

extern "C" void kernel_launch(void* const* d_in, const int* in_sizes, int n_in,
                              void* d_out, int out_size, void* d_ws,
                              size_t ws_size, hipStream_t stream) {
  const float* x = (const float*)d_in[0];
  float* out = (float*)d_out;
  const long long n = (long long)in_sizes[0];

  const long long n4 = n >> 2;
  const long long rem = n - (n4 << 2);

  if (n4 > 0) {
    const int threads = 256;  // 8 wave32 per block; fills a WGP twice
    // Fast path if the vec4 byte extent fits 32-bit signed offsets.
    if (n4 * 16 < (1LL << 31)) {
      int blocks = (int)((n4 + threads - 1) / threads);
      ceil_route_vec4_fast<<<blocks, threads, 0, stream>>>(
          (const float4v*)x, (float4v*)out, (int)n4);
    } else {
      long long blocks = (n4 + threads - 1) / threads;
      if (blocks > (1LL << 22)) blocks = (1LL << 22);
      ceil_route_vec4_generic<<<(dim3)(unsigned)blocks, threads, 0, stream>>>(
          (const float4v*)x, (float4v*)out, n4);
    }
  }
  if (rem > 0) {
    ceil_route_tail<<<1, 64, 0, stream>>>(x, out, n4 << 2, n);
  }
}